// MetaRas_54176717471778
// MI455X (gfx1250) — compile-verified
//
#include <hip/hip_runtime.h>
#include <hip/hip_bf16.h>
#include <stdint.h>

#define NFACES   512
#define FSTRIDE  28          // floats per face record in LDS (112 bytes, 16B aligned)
#define IMG      256
#define NPIX     (IMG*IMG)

__device__ __forceinline__ float fast_rcp(float x) {
#if __has_builtin(__builtin_amdgcn_rcpf)
    return __builtin_amdgcn_rcpf(x);
#else
    return 1.0f / x;
#endif
}
__device__ __forceinline__ float fast_exp2(float x) {
#if __has_builtin(__builtin_amdgcn_exp2f)
    return __builtin_amdgcn_exp2f(x);
#else
    return exp2f(x);
#endif
}
__device__ __forceinline__ float fast_sqrt(float x) {
#if __has_builtin(__builtin_amdgcn_sqrtf)
    return __builtin_amdgcn_sqrtf(x);
#else
    return sqrtf(x);
#endif
}
__device__ __forceinline__ float clamp01(float x) {
    return fminf(fmaxf(x, 0.0f), 1.0f);
}

// point-to-segment squared distance; (rx,ry) = p - a, (dx,dy) = b - a, il = 1/max(|d|^2,eps)
__device__ __forceinline__ float edge_d2(float rx, float ry, float dx, float dy, float il) {
    float t = clamp01((rx * dx + ry * dy) * il);
    float ex = rx - t * dx;
    float ey = ry - t * dy;
    return ex * ex + ey * ey;
}

__global__ __launch_bounds__(256)
void MetaRas_kernel(const float* __restrict__ verts,   // (1,512,3,3)
                    const float* __restrict__ tex,     // (1,512,3,3)
                    float* __restrict__ out)           // (1,4,256,256)
{
    __shared__ float lds[NFACES * FSTRIDE];            // 57,344 bytes

    const int t = threadIdx.x;

    // ---------------- Stage face_textures (512*9 fp32) into LDS via async DMA ----------
    // Each lane issues 18 global_load_async_to_lds_b32 with a scattered per-lane LDS
    // destination (face*FSTRIDE + 18 + k). Tracked by ASYNCcnt.
    {
        #pragma unroll
        for (int i = 0; i < 18; ++i) {
            int idx = t + i * 256;               // 0..4607
            int f   = idx / 9;
            int k   = idx - f * 9;
            const float* gp = tex + idx;
            // flat shared pointer low 32 bits == wave-relative LDS byte address (ISA 10.2)
            uint32_t ldsa = (uint32_t)(uintptr_t)(lds + f * FSTRIDE + 18 + k);
            asm volatile("global_load_async_to_lds_b32 %0, %1, off"
                         :: "v"(ldsa), "v"(gp) : "memory");
        }
    }

    // ---------------- Precompute per-face derived constants into LDS ------------------
    // (overlaps with the async texture DMA)
    #pragma unroll
    for (int j = 0; j < 2; ++j) {
        int f = t * 2 + j;
        const float* gv = verts + f * 9;
        float x0 = gv[0], y0 = gv[1], z0 = gv[2];
        float x1 = gv[3], y1 = gv[4], z1 = gv[5];
        float x2 = gv[6], y2 = gv[7], z2 = gv[8];

        float den = (y1 - y2) * (x0 - x2) + (x2 - x1) * (y0 - y2);
        den = (fabsf(den) < 1e-10f) ? 1e-10f : den;
        float invden = fast_rcp(den);

        float A0 = (y1 - y2) * invden, B0 = (x2 - x1) * invden;
        float C0 = -(A0 * x2 + B0 * y2);
        float A1 = (y2 - y0) * invden, B1 = (x0 - x2) * invden;
        float C1 = -(A1 * x2 + B1 * y2);

        float dx01 = x1 - x0, dy01 = y1 - y0;
        float dx12 = x2 - x1, dy12 = y2 - y1;
        float dx20 = x0 - x2, dy20 = y0 - y2;

        float* d = lds + f * FSTRIDE;
        d[0]  = A0;  d[1]  = B0;  d[2]  = C0;  d[3]  = A1;
        d[4]  = B1;  d[5]  = C1;  d[6]  = x0;  d[7]  = y0;
        d[8]  = x1;  d[9]  = y1;  d[10] = x2;  d[11] = y2;
        d[12] = fast_rcp(fmaxf(dx01 * dx01 + dy01 * dy01, 1e-12f));
        d[13] = fast_rcp(fmaxf(dx12 * dx12 + dy12 * dy12, 1e-12f));
        d[14] = fast_rcp(fmaxf(dx20 * dx20 + dy20 * dy20, 1e-12f));
        d[15] = fast_rcp(z0);
        d[16] = fast_rcp(z1);
        d[17] = fast_rcp(z2);
        d[27] = 0.0f;                              // pad (slots 18..26 filled by async DMA)
    }

    // all async DMA of this wave complete, then workgroup-visible
#if __has_builtin(__builtin_amdgcn_s_wait_asynccnt)
    __builtin_amdgcn_s_wait_asynccnt(0);
#else
    asm volatile("s_wait_asynccnt 0" ::: "memory");
#endif
    __syncthreads();

    // ---------------- Per-pixel single-pass online softmax over 512 faces -------------
    const int p = blockIdx.x * 256 + t;            // pixel id
    const int h = p >> 8, w = p & 255;
    const float px = ((float)w + 0.5f) * (2.0f / 256.0f) - 1.0f;
    const float py = ((float)h + 0.5f) * (2.0f / 256.0f) - 1.0f;

    const float K2 = 1442.6950408889634f;          // log2(e)/GAMMA
    float m  = 1e-3f;                              // running max of zn, seeded with EPS
    float S  = 0.0f;                               // running sum of wexp
    float R0 = 0.0f, R1 = 0.0f, R2 = 0.0f;         // running rgb numerators
    float keep = 1.0f;                             // prod(1 - prob)

    for (int f = 0; f < NFACES; ++f) {
        const float4* q = (const float4*)(lds + f * FSTRIDE);  // broadcast ds_load_b128 x7
        float4 q0 = q[0], q1 = q[1], q2 = q[2], q3 = q[3], q4 = q[4], q5 = q[5], q6 = q[6];

        // barycentrics (affine in px,py)
        float w0 = fmaf(q0.x, px, fmaf(q0.y, py, q0.z));
        float w1 = fmaf(q0.w, px, fmaf(q1.x, py, q1.y));
        float w2 = 1.0f - w0 - w1;
        bool inside = (w0 >= 0.0f) & (w1 >= 0.0f) & (w2 >= 0.0f);

        // min squared distance to the 3 edges
        float x0 = q1.z, y0 = q1.w, x1 = q2.x, y1 = q2.y, x2 = q2.z, y2 = q2.w;
        float d2a = edge_d2(px - x0, py - y0, x1 - x0, y1 - y0, q3.x);
        float d2b = edge_d2(px - x1, py - y1, x2 - x1, y2 - y1, q3.y);
        float d2c = edge_d2(px - x2, py - y2, x0 - x2, y0 - y2, q3.z);
        float dist = fast_sqrt(fminf(fminf(d2a, d2b), d2c) + 1e-12f);

        // prob = sigmoid(sgn*dist/SIGMA)
        float arg  = inside ? (dist * 100.0f) : (-dist * 100.0f);
        float prob = fast_rcp(1.0f + fast_exp2(arg * -1.4426950408889634f));

        // clipped, normalized barycentrics; interpolated depth
        float c0 = clamp01(w0), c1 = clamp01(w1), c2 = clamp01(w2);
        float invs = fast_rcp(fmaxf(c0 + c1 + c2, 1e-12f));
        c0 *= invs; c1 *= invs; c2 *= invs;
        float zpi = fmaf(c0, q3.w, fmaf(c1, q4.x, c2 * q4.y));
        float zp  = fast_rcp(fmaxf(zpi, 1e-12f));

        float vmask = ((zp > 1.0f) & (zp < 100.0f)) ? 1.0f : 0.0f;
        prob *= vmask;
        float zn = clamp01((100.0f - zp) * (1.0f / 99.0f)) * vmask;

        // online softmax update (branchless rescale)
        float m2    = fmaxf(m, zn);
        float scale = fast_exp2((m - m2) * K2);
        float e     = prob * fast_exp2((zn - m2) * K2);

        float col0 = fmaf(c0, q4.z, fmaf(c1, q5.y, c2 * q6.x));   // R
        float col1 = fmaf(c0, q4.w, fmaf(c1, q5.z, c2 * q6.y));   // G
        float col2 = fmaf(c0, q5.x, fmaf(c1, q5.w, c2 * q6.z));   // B

        S  = fmaf(S,  scale, e);
        R0 = fmaf(R0, scale, e * col0);
        R1 = fmaf(R1, scale, e * col1);
        R2 = fmaf(R2, scale, e * col2);
        m  = m2;
        keep *= (1.0f - prob);
    }

    // finalize: m == max(max_f zn, EPS) because m was seeded with EPS
    float wbg  = fast_exp2((1e-3f - m) * K2);
    float invd = fast_rcp(S + wbg);                // BG is (0,0,0)

    out[p]              = R0 * invd;
    out[NPIX + p]       = R1 * invd;
    out[2 * NPIX + p]   = R2 * invd;
    out[3 * NPIX + p]   = 1.0f - keep;
}

extern "C" void kernel_launch(void* const* d_in, const int* in_sizes, int n_in,
                              void* d_out, int out_size, void* d_ws, size_t ws_size,
                              hipStream_t stream) {
    (void)in_sizes; (void)n_in; (void)d_ws; (void)ws_size; (void)out_size;
    const float* verts = (const float*)d_in[0];   // (1,512,3,3) fp32
    const float* tex   = (const float*)d_in[1];   // (1,512,3,3) fp32
    float* out = (float*)d_out;                   // (1,4,256,256) fp32
    MetaRas_kernel<<<NPIX / 256, 256, 0, stream>>>(verts, tex, out);
}